// OneLayer_RNN_60902636257618
// MI455X (gfx1250) — compile-verified
//
#include <hip/hip_runtime.h>
#include <math.h>

// Problem sizes (match reference): L=1024 timesteps, B=64 batch, C=512 in, H=1024 hidden
constexpr int Lq = 1024;
constexpr int Bq = 64;
constexpr int Cq = 512;
constexpr int Hq = 1024;

typedef float v2f __attribute__((ext_vector_type(2)));
typedef float v8f __attribute__((ext_vector_type(8)));

// ---------------------------------------------------------------------------
// Phase 1: xW = x @ W_ih + b   -> written into outputs region of d_out
// grid (M/64 = 1024, N/64 = 16), 256 threads (8 waves), LDS-staged fp32 WMMA
// ---------------------------------------------------------------------------
__global__ __launch_bounds__(256)
void rnn_xproj(const float* __restrict__ x, const float* __restrict__ Wih,
               const float* __restrict__ bias, float* __restrict__ out,
               float* __restrict__ h0, unsigned* __restrict__ cnt)
{
  constexpr int AS = 36;  // A panel stride (floats): 36*r mod 64 distinct, 16B-aligned rows
  constexpr int BS = 68;  // B panel stride (floats): 16B-aligned rows
  __shared__ float As[64 * AS];
  __shared__ float Bs[32 * BS];

  const int tid = threadIdx.x;

  // One block initializes scan state h0 = 0 and the grid-barrier counter.
  if (blockIdx.x == 0 && blockIdx.y == 0) {
    for (int i = tid; i < Bq * Hq; i += 256) h0[i] = 0.0f;
    if (tid == 0) *cnt = 0u;
  }

  const int m_base = blockIdx.x * 64;
  const int n_base = blockIdx.y * 64;
  const int wave = tid >> 5;          // wave32
  const int lane = tid & 31;
  const int mt  = wave & 3;           // M-tile 0..3
  const int nt0 = (wave >> 2) * 2;    // N-tile pair
  const int nt1 = nt0 + 1;
  const int lrow = lane & 15;         // row/col within 16
  const int lk   = (lane >> 4) << 1;  // K sub-offset: 0 or 2

  v8f c0 = {};
  v8f c1 = {};

  for (int k0 = 0; k0 < Cq; k0 += 32) {
    __syncthreads();
    // Stage A: 64x32 tile of x (row-major [M][C])
#pragma unroll
    for (int j = 0; j < 2; ++j) {
      int i = tid + j * 256;                 // 512 float4
      int r = i >> 3, c4 = (i & 7) << 2;
      float4 v = *(const float4*)(x + (size_t)(m_base + r) * Cq + k0 + c4);
      *(float4*)(As + r * AS + c4) = v;
    }
    // Stage B: 32x64 tile of W_ih (row-major [C][H])
#pragma unroll
    for (int j = 0; j < 2; ++j) {
      int i = tid + j * 256;                 // 512 float4
      int r = i >> 4, c4 = (i & 15) << 2;
      float4 v = *(const float4*)(Wih + (size_t)(k0 + r) * Hq + n_base + c4);
      *(float4*)(Bs + r * BS + c4) = v;
    }
    __syncthreads();

    const float* arow  = As + (mt * 16 + lrow) * AS + lk;
    const float* bcol0 = Bs + lk * BS + nt0 * 16 + lrow;
    const float* bcol1 = Bs + lk * BS + nt1 * 16 + lrow;
#pragma unroll
    for (int kk = 0; kk < 32; kk += 4) {
      v2f a;  a.x  = arow[kk];        a.y  = arow[kk + 1];
      v2f b0; b0.x = bcol0[kk * BS];  b0.y = bcol0[kk * BS + BS];
      v2f b1; b1.x = bcol1[kk * BS];  b1.y = bcol1[kk * BS + BS];
      c0 = __builtin_amdgcn_wmma_f32_16x16x4_f32(false, a, false, b0, (short)0, c0, false, false);
      c1 = __builtin_amdgcn_wmma_f32_16x16x4_f32(false, a, false, b1, (short)0, c1, false, false);
    }
  }

  // Epilogue: add bias, store C tiles (VGPR i: lanes 0-15 -> M=i, lanes 16-31 -> M=i+8)
  const int gcol0 = n_base + nt0 * 16 + lrow;
  const int gcol1 = n_base + nt1 * 16 + lrow;
  const float bb0 = bias[gcol0];
  const float bb1 = bias[gcol1];
  const int rbase = m_base + mt * 16 + ((lane >> 4) << 3);
#pragma unroll
  for (int i = 0; i < 8; ++i) {
    const size_t r = (size_t)(rbase + i);
    out[r * Hq + gcol0] = c0[i] + bb0;
    out[r * Hq + gcol1] = c1[i] + bb1;
  }
}

// ---------------------------------------------------------------------------
// Phase 2: persistent sequential scan: h = tanh(xW[t] + h @ W_hh)
// 64 blocks (16 columns of H each), 128 threads (4 waves = 4 M-tiles of 16).
// W_hh column slice staged once in LDS (64KB) via gfx1250 async global->LDS
// copies (ASYNCcnt); h double-buffered in d_ws; device-wide barrier via a
// monotonic agent-scope atomic counter. K loop uses 4 independent WMMA
// accumulators to break the SRC2 dependency chain.
// ---------------------------------------------------------------------------
__global__ __launch_bounds__(128)
void rnn_scan(const float* __restrict__ Whh, float* __restrict__ out,
              float* __restrict__ hA, float* __restrict__ hB,
              unsigned* __restrict__ cnt, int nblocks)
{
  __shared__ float Ws[Hq * 16];   // 64 KB: W_hh[:, n0:n0+16]

  const int tid = threadIdx.x;
  const int n0  = blockIdx.x * 16;

  // Stage the constant W_hh column slice once, using async global->LDS B128
  // copies. INST_OFFSET is added to both the LDS and global addresses, so one
  // (lds,global) address pair covers a 16-float row with 4 offsets.
  {
    const unsigned lds_base = (unsigned)(uintptr_t)(&Ws[0]);
    for (int k = tid; k < Hq; k += 128) {
      const unsigned ldsa = lds_base + (unsigned)(k * 16 * 4);
      const float*   src  = Whh + (size_t)k * Hq + n0;
      asm volatile("global_load_async_to_lds_b128 %0, %1, off offset:0"
                   :: "v"(ldsa), "v"(src) : "memory");
      asm volatile("global_load_async_to_lds_b128 %0, %1, off offset:16"
                   :: "v"(ldsa), "v"(src) : "memory");
      asm volatile("global_load_async_to_lds_b128 %0, %1, off offset:32"
                   :: "v"(ldsa), "v"(src) : "memory");
      asm volatile("global_load_async_to_lds_b128 %0, %1, off offset:48"
                   :: "v"(ldsa), "v"(src) : "memory");
    }
    asm volatile("s_wait_asynccnt 0x0" ::: "memory");
  }
  __syncthreads();

  const int wave = tid >> 5;          // 0..3 -> M-tile
  const int lane = tid & 31;
  const int lrow = lane & 15;
  const int lk   = (lane >> 4) << 1;  // 0 or 2
  const int m0   = wave * 16;

  // B operand base: b.x = Ws[(k+lk)*16 + lrow], b.y one K-row further
  const float* bbase = Ws + lk * 16 + lrow;

  // Epilogue element mapping (VGPR i: lanes 0-15 -> M=i, lanes 16-31 -> M=i+8)
  const int col = n0 + lrow;
  const int rb  = m0 + ((lane >> 4) << 3);

  for (int t = 0; t < Lq; ++t) {
    const float* hc = (t & 1) ? hB : hA;
    float*       hn = (t & 1) ? hA : hB;
    const float* arow = hc + (size_t)(m0 + lrow) * Hq + lk;
    float* outT = out + (size_t)t * (Bq * Hq);

    // Preload xW[t] slice early so its L2 latency hides under the K loop.
    float xw[8];
#pragma unroll
    for (int i = 0; i < 8; ++i)
      xw[i] = outT[(size_t)(rb + i) * Hq + col];

    // K = 1024 reduction, 4 independent accumulators (breaks WMMA SRC2 chain)
    v8f acc0 = {}, acc1 = {}, acc2 = {}, acc3 = {};
#pragma unroll 4
    for (int k = 0; k < Hq; k += 16) {
      v2f a0; a0.x = arow[k];      a0.y = arow[k + 1];
      v2f b0; b0.x = bbase[k * 16];        b0.y = bbase[k * 16 + 16];
      acc0 = __builtin_amdgcn_wmma_f32_16x16x4_f32(false, a0, false, b0, (short)0, acc0, false, false);

      v2f a1; a1.x = arow[k + 4];  a1.y = arow[k + 5];
      v2f b1; b1.x = bbase[(k + 4) * 16];  b1.y = bbase[(k + 4) * 16 + 16];
      acc1 = __builtin_amdgcn_wmma_f32_16x16x4_f32(false, a1, false, b1, (short)0, acc1, false, false);

      v2f a2; a2.x = arow[k + 8];  a2.y = arow[k + 9];
      v2f b2; b2.x = bbase[(k + 8) * 16];  b2.y = bbase[(k + 8) * 16 + 16];
      acc2 = __builtin_amdgcn_wmma_f32_16x16x4_f32(false, a2, false, b2, (short)0, acc2, false, false);

      v2f a3; a3.x = arow[k + 12]; a3.y = arow[k + 13];
      v2f b3; b3.x = bbase[(k + 12) * 16]; b3.y = bbase[(k + 12) * 16 + 16];
      acc3 = __builtin_amdgcn_wmma_f32_16x16x4_f32(false, a3, false, b3, (short)0, acc3, false, false);
    }
    const v8f acc = (acc0 + acc1) + (acc2 + acc3);

    // Epilogue: pre-activation = xW[t] + acc; tanh; write outputs/state
#pragma unroll
    for (int i = 0; i < 8; ++i) {
      const size_t idx = (size_t)(rb + i) * Hq + col;
      float v = tanhf(xw[i] + acc[i]);
      outT[idx] = v;        // outputs[t]
      hn[idx]   = v;        // next state
      if (t == Lq - 1)
        out[(size_t)Lq * Bq * Hq + idx] = v;   // final_state tail of d_out
    }

    // ---- device-wide barrier (monotonic counter, zeroed by phase 1) ----
    __threadfence();
    __syncthreads();
    if (tid == 0) {
      __hip_atomic_fetch_add(cnt, 1u, __ATOMIC_RELEASE, __HIP_MEMORY_SCOPE_AGENT);
      const unsigned target = (unsigned)(t + 1) * (unsigned)nblocks;
      while (__hip_atomic_load(cnt, __ATOMIC_ACQUIRE, __HIP_MEMORY_SCOPE_AGENT) < target)
        __builtin_amdgcn_s_sleep(2);
    }
    __syncthreads();
  }
}

// ---------------------------------------------------------------------------
// Host launch
// ---------------------------------------------------------------------------
extern "C" void kernel_launch(void* const* d_in, const int* in_sizes, int n_in,
                              void* d_out, int out_size, void* d_ws, size_t ws_size,
                              hipStream_t stream) {
  const float* x    = (const float*)d_in[0];   // [L,B,C]
  const float* Wih  = (const float*)d_in[1];   // [C,H]
  const float* Whh  = (const float*)d_in[2];   // [H,H]
  const float* bias = (const float*)d_in[3];   // [H]
  float* out = (float*)d_out;                  // outputs [L,B,H] ++ final [B,H]

  unsigned* cnt = (unsigned*)d_ws;
  float* hA = (float*)((char*)d_ws + 256);
  float* hB = hA + Bq * Hq;                    // ws usage: 256B + 2*256KB

  dim3 g1((Lq * Bq) / 64, Hq / 64);            // (1024, 16)
  rnn_xproj<<<g1, dim3(256), 0, stream>>>(x, Wih, bias, out, hA, cnt);

  const int nb = Hq / 16;                      // 64 blocks
  rnn_scan<<<dim3(nb), dim3(128), 0, stream>>>(Whh, out, hA, hB, cnt, nb);
}